// WeightedRigidAlign_23124103922401
// MI455X (gfx1250) — compile-verified
//
#include <hip/hip_runtime.h>
#include <cmath>

typedef __attribute__((ext_vector_type(2))) float v2f;
typedef __attribute__((ext_vector_type(8))) float v8f;
// 4-byte-aligned 2-float vector for dword-aligned b64 loads
typedef float v2f_u __attribute__((ext_vector_type(2), aligned(4)));

#define THREADS 256

// ---------------------------------------------------------------------------
// Kernel 1: per-batch 4x4 weighted moment matrix.
// acc[0..8]  = sum w * t_i * p_j   (3x3 row-major)
// acc[9..11] = sum w * t_i
// acc[12..14]= sum w * p_j
// acc[15]    = sum w
// ---------------------------------------------------------------------------
__global__ void wra_moments(const float* __restrict__ pred,
                            const float* __restrict__ truc,
                            const float* __restrict__ wts,
                            float* __restrict__ mom, int N) {
  const int b = blockIdx.x;
  const int tid = threadIdx.x;
  const size_t base = (size_t)b * N;

  float acc[16];
#pragma unroll
  for (int i = 0; i < 16; ++i) acc[i] = 0.f;

  for (int p = tid; p < N; p += THREADS) {
    const size_t idx = base + p;
    const float* tp = truc + idx * 3;
    const float* pp = pred + idx * 3;
    if (p + THREADS < N) {
      __builtin_prefetch(tp + THREADS * 3, 0, 0);
      __builtin_prefetch(pp + THREADS * 3, 0, 0);
    }
    const float w  = wts[idx];
    const float t0 = tp[0], t1 = tp[1], t2 = tp[2];
    const float p0 = pp[0], p1 = pp[1], p2 = pp[2];
    const float wt0 = w * t0, wt1 = w * t1, wt2 = w * t2;
    acc[0]  += wt0 * p0; acc[1]  += wt0 * p1; acc[2]  += wt0 * p2;
    acc[3]  += wt1 * p0; acc[4]  += wt1 * p1; acc[5]  += wt1 * p2;
    acc[6]  += wt2 * p0; acc[7]  += wt2 * p1; acc[8]  += wt2 * p2;
    acc[9]  += wt0;      acc[10] += wt1;      acc[11] += wt2;
    acc[12] += w * p0;   acc[13] += w * p1;   acc[14] += w * p2;
    acc[15] += w;
  }

  // wave32 reduction
#pragma unroll
  for (int off = 16; off > 0; off >>= 1) {
#pragma unroll
    for (int i = 0; i < 16; ++i) acc[i] += __shfl_down(acc[i], off, 32);
  }

  __shared__ float red[THREADS / 32][16];
  const int wave = tid >> 5;
  const int lane = tid & 31;
  if (lane == 0) {
#pragma unroll
    for (int i = 0; i < 16; ++i) red[wave][i] = acc[i];
  }
  __syncthreads();
  if (tid < 16) {
    float s = 0.f;
#pragma unroll
    for (int w = 0; w < THREADS / 32; ++w) s += red[w][tid];
    mom[(size_t)b * 16 + tid] = s;
  }
}

// ---------------------------------------------------------------------------
// Kernel 2: per-batch 3x3 Kabsch. One thread per batch.
// Produces 3x4 affine T (row-major, rows j=0..2):
//   out_j = sum_k pred_k * T[j][k]  (k<3)  + T[j][3]
// with T[j][k] = R[k][j], T[j][3] = tc[j] - sum_k R[k][j]*pc[k]
// ---------------------------------------------------------------------------
__global__ void wra_kabsch(const float* __restrict__ mom,
                           float* __restrict__ Tf, int B) {
  const int b = blockIdx.x * blockDim.x + threadIdx.x;
  if (b >= B) return;
  const float* m = mom + (size_t)b * 16;

  const float sw  = m[15];
  const float inv = 1.0f / sw;
  const float st[3] = {m[9],  m[10], m[11]};
  const float sp[3] = {m[12], m[13], m[14]};

  // cov[i][j] = sum w*(t_i-tc_i)*(p_j-pc_j) = S_tp[i][j] - st[i]*sp[j]/sw
  float A[3][3];
#pragma unroll
  for (int i = 0; i < 3; ++i)
#pragma unroll
    for (int j = 0; j < 3; ++j)
      A[i][j] = m[i * 3 + j] - st[i] * sp[j] * inv;

  // Bm = A^T A (symmetric)
  float Bm[3][3];
#pragma unroll
  for (int i = 0; i < 3; ++i)
#pragma unroll
    for (int j = 0; j < 3; ++j)
      Bm[i][j] = A[0][i] * A[0][j] + A[1][i] * A[1][j] + A[2][i] * A[2][j];

  float V[3][3] = {{1.f, 0.f, 0.f}, {0.f, 1.f, 0.f}, {0.f, 0.f, 1.f}};

  // cyclic Jacobi eigendecomposition of Bm
  for (int sweep = 0; sweep < 10; ++sweep) {
#pragma unroll
    for (int pair = 0; pair < 3; ++pair) {
      const int p = (pair == 2) ? 1 : 0;
      const int q = (pair == 0) ? 1 : 2;
      const float apq = Bm[p][q];
      if (fabsf(apq) > 1e-30f) {
        const float app = Bm[p][p], aqq = Bm[q][q];
        const float tau = (aqq - app) / (2.0f * apq);
        const float t = (tau >= 0.f ? 1.f : -1.f) /
                        (fabsf(tau) + sqrtf(1.0f + tau * tau));
        const float c = 1.0f / sqrtf(1.0f + t * t);
        const float s = t * c;
#pragma unroll
        for (int k = 0; k < 3; ++k) {  // Bm = Bm * J
          const float bkp = Bm[k][p], bkq = Bm[k][q];
          Bm[k][p] = c * bkp - s * bkq;
          Bm[k][q] = s * bkp + c * bkq;
        }
#pragma unroll
        for (int k = 0; k < 3; ++k) {  // Bm = J^T * Bm
          const float bpk = Bm[p][k], bqk = Bm[q][k];
          Bm[p][k] = c * bpk - s * bqk;
          Bm[q][k] = s * bpk + c * bqk;
        }
#pragma unroll
        for (int k = 0; k < 3; ++k) {  // V = V * J
          const float vkp = V[k][p], vkq = V[k][q];
          V[k][p] = c * vkp - s * vkq;
          V[k][q] = s * vkp + c * vkq;
        }
      }
    }
  }

  float ev[3] = {Bm[0][0], Bm[1][1], Bm[2][2]};
  // sort eigenpairs descending (matches jnp.linalg.svd ordering)
#pragma unroll
  for (int pass = 0; pass < 2; ++pass) {
#pragma unroll
    for (int i = 0; i < 2; ++i) {
      if (ev[i] < ev[i + 1]) {
        const float e = ev[i]; ev[i] = ev[i + 1]; ev[i + 1] = e;
#pragma unroll
        for (int k = 0; k < 3; ++k) {
          const float v = V[k][i]; V[k][i] = V[k][i + 1]; V[k][i + 1] = v;
        }
      }
    }
  }

  // U columns: u_c = A * v_c / sigma_c, with degenerate fallbacks
  float U[3][3];
#pragma unroll
  for (int c = 0; c < 3; ++c) {
    float u0 = A[0][0] * V[0][c] + A[0][1] * V[1][c] + A[0][2] * V[2][c];
    float u1 = A[1][0] * V[0][c] + A[1][1] * V[1][c] + A[1][2] * V[2][c];
    float u2 = A[2][0] * V[0][c] + A[2][1] * V[1][c] + A[2][2] * V[2][c];
    const float nrm = sqrtf(u0 * u0 + u1 * u1 + u2 * u2);
    if (nrm > 1e-12f) {
      const float r = 1.0f / nrm;
      U[0][c] = u0 * r; U[1][c] = u1 * r; U[2][c] = u2 * r;
    } else if (c == 0) {
      U[0][0] = 1.f; U[1][0] = 0.f; U[2][0] = 0.f;
    } else if (c == 1) {
      // any unit vector orthogonal to column 0
      float ax = fabsf(U[0][0]), ay = fabsf(U[1][0]), az = fabsf(U[2][0]);
      float e0 = 0.f, e1 = 0.f, e2 = 0.f;
      if (ax <= ay && ax <= az) e0 = 1.f; else if (ay <= az) e1 = 1.f; else e2 = 1.f;
      float c0 = e1 * U[2][0] - e2 * U[1][0];
      float c1 = e2 * U[0][0] - e0 * U[2][0];
      float c2 = e0 * U[1][0] - e1 * U[0][0];
      const float r = 1.0f / sqrtf(c0 * c0 + c1 * c1 + c2 * c2 + 1e-30f);
      U[0][1] = c0 * r; U[1][1] = c1 * r; U[2][1] = c2 * r;
    } else {
      // cross of first two columns
      float c0 = U[1][0] * U[2][1] - U[2][0] * U[1][1];
      float c1 = U[2][0] * U[0][1] - U[0][0] * U[2][1];
      float c2 = U[0][0] * U[1][1] - U[1][0] * U[0][1];
      const float r = 1.0f / sqrtf(c0 * c0 + c1 * c1 + c2 * c2 + 1e-30f);
      U[0][2] = c0 * r; U[1][2] = c1 * r; U[2][2] = c2 * r;
    }
  }

  // rot = U @ V (reference: V = Vh^T, rot = matmul(U, V))
  float R[3][3];
#pragma unroll
  for (int i = 0; i < 3; ++i)
#pragma unroll
    for (int j = 0; j < 3; ++j)
      R[i][j] = U[i][0] * V[0][j] + U[i][1] * V[1][j] + U[i][2] * V[2][j];

  const float det =
      R[0][0] * (R[1][1] * R[2][2] - R[1][2] * R[2][1]) -
      R[0][1] * (R[1][0] * R[2][2] - R[1][2] * R[2][0]) +
      R[0][2] * (R[1][0] * R[2][1] - R[1][1] * R[2][0]);
  if (det < 0.f) {  // flipping V's last column == negating rot's last column
    R[0][2] = -R[0][2]; R[1][2] = -R[1][2]; R[2][2] = -R[2][2];
  }

  const float pc[3] = {sp[0] * inv, sp[1] * inv, sp[2] * inv};
  const float tc[3] = {st[0] * inv, st[1] * inv, st[2] * inv};
  float* T = Tf + (size_t)b * 12;
#pragma unroll
  for (int j = 0; j < 3; ++j) {
    T[j * 4 + 0] = R[0][j];
    T[j * 4 + 1] = R[1][j];
    T[j * 4 + 2] = R[2][j];
    T[j * 4 + 3] = tc[j] - (R[0][j] * pc[0] + R[1][j] * pc[1] + R[2][j] * pc[2]);
  }
}

// ---------------------------------------------------------------------------
// Kernel 3: apply affine transform with v_wmma_f32_16x16x4_f32.
// D(16x16) = A(16x4) x B(4x16):
//   A rows 0..2 = affine rows T[j][0..3]      (lanes 0-15: K0,K1; 16-31: K2,K3)
//   B cols n    = point n's (x, y, z, 1)      (lanes 0-15: K0,K1; 16-31: K2,K3)
// Branchless B build: every lane loads b64 at element offset point*3 + (lo?0:1)
//   lo lanes get (x,y) -> (K0,K1); hi lanes get (y,z) -> select z, append 1.0.
// Loads are batched 8-deep ahead of the WMMA+store sections so the wave keeps
// ~8 global_load_b64 in flight (bandwidth-bound pass; hide HBM latency).
// ---------------------------------------------------------------------------
__global__ void wra_apply(const float* __restrict__ pred,
                          const float* __restrict__ Tf,
                          float* __restrict__ out, int N) {
  const int b = blockIdx.x;
  const int tid = threadIdx.x;
  const int wave = tid >> 5;
  const int lane = tid & 31;
  const int m = lane & 15;
  const bool lo = lane < 16;

  const float* T = Tf + (size_t)b * 12;
  v2f a;
  a.x = 0.f; a.y = 0.f;
  if (m < 3) {
    a.x = lo ? T[m * 4 + 0] : T[m * 4 + 2];
    a.y = lo ? T[m * 4 + 1] : T[m * 4 + 3];
  }

  const size_t base = (size_t)b * N;
  const int p0 = wave * 16 + m;                       // first point for this lane
  const int stride = (THREADS / 32) * 16;             // 128 points / block iter
  const int estep = stride * 3;                       // element step per iter

  const float* lp = pred + (base + p0) * 3 + (lo ? 0 : 1);
  float*       op = out  + (base + p0) * 3;

  const int iters = N / stride;                       // 8192/128 = 64
  const int BATCH = 8;

  for (int it = 0; it < iters; it += BATCH) {
    // Issue all BATCH loads first -> 8 outstanding global_load_b64 per wave.
    v2f_u vv[BATCH];
#pragma unroll
    for (int u = 0; u < BATCH; ++u)
      vv[u] = *(const v2f_u*)(lp + u * estep);

#pragma unroll
    for (int u = 0; u < BATCH; ++u) {
      v2f bm;
      bm.x = lo ? vv[u].x : vv[u].y;                  // lo: x   | hi: z
      bm.y = lo ? vv[u].y : 1.0f;                     // lo: y   | hi: 1

      v8f c = {0.f, 0.f, 0.f, 0.f, 0.f, 0.f, 0.f, 0.f};
      v8f d = __builtin_amdgcn_wmma_f32_16x16x4_f32(
          /*neg_a=*/false, a, /*neg_b=*/false, bm,
          /*c_mod=*/(short)0, c, /*reuse_a=*/false, /*reuse_b=*/false);

      if (lo) {                                       // rows 0..2 live in lanes 0-15
        float* o = op + u * estep;
        o[0] = d[0];  // x'
        o[1] = d[1];  // y'
        o[2] = d[2];  // z'
      }
    }
    lp += BATCH * estep;
    op += BATCH * estep;
  }
}

// ---------------------------------------------------------------------------
extern "C" void kernel_launch(void* const* d_in, const int* in_sizes, int n_in,
                              void* d_out, int out_size, void* d_ws, size_t ws_size,
                              hipStream_t stream) {
  const float* pred = (const float*)d_in[0];
  const float* truc = (const float*)d_in[1];
  const float* wts  = (const float*)d_in[2];
  float* out = (float*)d_out;

  const int N = 8192;
  const int B = in_sizes[2] / N;  // weights is (B, N)

  float* mom = (float*)d_ws;                 // B * 16 floats
  float* T   = mom + (size_t)B * 16;         // B * 12 floats

  wra_moments<<<B, THREADS, 0, stream>>>(pred, truc, wts, mom, N);
  wra_kabsch<<<(B + 255) / 256, 256, 0, stream>>>(mom, T, B);
  wra_apply<<<B, THREADS, 0, stream>>>(pred, T, out, N);
}